// MocoModel_3032246911264
// MI455X (gfx1250) — compile-verified
//
#include <hip/hip_runtime.h>
#include <hip/hip_bf16.h>

#define NN    50000
#define EE    800000
#define QQ    256
#define D_IN  128
#define D_H   256
#define D_OUT 128
#define EPSV  1e-5f

typedef float v2f __attribute__((ext_vector_type(2)));
typedef float v8f __attribute__((ext_vector_type(8)));

#define WMMA_F32(A, B, C) \
    __builtin_amdgcn_wmma_f32_16x16x4_f32(false, (A), false, (B), (short)0, (C), false, false)

// ---------------------------------------------------------------- small utils
__global__ void k_fill_f32(float* p, float v, int n) {
    int i = blockIdx.x * blockDim.x + threadIdx.x;
    if (i < n) p[i] = v;
}
__global__ void k_zero_i32(int* p, int n) {
    int i = blockIdx.x * blockDim.x + threadIdx.x;
    if (i < n) p[i] = 0;
}

// ------------------------------------------------------------- degree / dinv
__global__ void k_deg_scatter(const int* __restrict__ col, const float* __restrict__ ew,
                              float* __restrict__ deg) {
    int e = blockIdx.x * blockDim.x + threadIdx.x;
    if (e < EE) atomicAdd(&deg[col[e]], ew[e]);
}
__global__ void k_dinv(const float* __restrict__ deg, float* __restrict__ dinv) {
    int i = blockIdx.x * blockDim.x + threadIdx.x;
    if (i < NN) dinv[i] = (deg[i] > 0.f) ? rsqrtf(deg[i]) : 0.f;
}

// --------------------------------------------------------------- WMMA GEMM
// D[m][n] = sum_k A[m][k] * W[n][k].
// Block = 8 waves: wave w -> m-tile (msuper*8+w), all waves share one 64-col
// n-chunk whose W rows are staged in LDS (row-padded to K+4 to spread banks).
// B fragments come from LDS (ds_load_b64, cheap dscnt waits); the single
// global A-fragment load is software-pipelined at distance 2.
__global__ void k_gemm_wmma(const float* __restrict__ A, const float* __restrict__ W,
                            float* __restrict__ D, int Mtiles, int K, int NC) {
    extern __shared__ float smem[];             // [64][K+4]
    const int tilesN = NC >> 6;
    const int nchunk = blockIdx.x % tilesN;
    const int msuper = blockIdx.x / tilesN;
    const int waveid = threadIdx.x >> 5;
    const int lane   = threadIdx.x & 31;
    const int mt     = msuper * 8 + waveid;
    const int nbase  = nchunk * 64;
    const int ldW    = K + 4;

    // cooperative, coalesced stage of W[nbase:nbase+64][0:K] into LDS
    for (int r = waveid; r < 64; r += 8) {
        const float4* src = (const float4*)(W + (size_t)(nbase + r) * K);
        float* dst = smem + r * ldW;
        for (int c4 = lane; c4 < (K >> 2); c4 += 32)
            *(float4*)(dst + c4 * 4) = src[c4];
    }
    __syncthreads();

    if (mt < Mtiles) {                          // wave-uniform guard: EXEC all-1
        const int half = lane >> 4, lr = lane & 15;
        const int k2 = 2 * half;
        const float* arow = A + (size_t)(mt * 16 + lr) * K + k2;
        const float* s0 = smem + ( 0 + lr) * ldW + k2;
        const float* s1 = smem + (16 + lr) * ldW + k2;
        const float* s2 = smem + (32 + lr) * ldW + k2;
        const float* s3 = smem + (48 + lr) * ldW + k2;

        v2f a0 = *(const v2f*)(arow);
        v2f a1 = *(const v2f*)(arow + 4);
        v8f acc0 = {}, acc1 = {}, acc2 = {}, acc3 = {};
        int k = 0;
        for (; k < K - 8; k += 4) {
            v2f a2 = *(const v2f*)(arow + k + 8);   // prefetch distance 2
            v2f b0 = *(const v2f*)(s0 + k);
            v2f b1 = *(const v2f*)(s1 + k);
            v2f b2 = *(const v2f*)(s2 + k);
            v2f b3 = *(const v2f*)(s3 + k);
            acc0 = WMMA_F32(a0, b0, acc0);
            acc1 = WMMA_F32(a0, b1, acc1);
            acc2 = WMMA_F32(a0, b2, acc2);
            acc3 = WMMA_F32(a0, b3, acc3);
            a0 = a1; a1 = a2;
        }
        for (; k < K; k += 4) {                     // final two k-steps
            v2f b0 = *(const v2f*)(s0 + k);
            v2f b1 = *(const v2f*)(s1 + k);
            v2f b2 = *(const v2f*)(s2 + k);
            v2f b3 = *(const v2f*)(s3 + k);
            acc0 = WMMA_F32(a0, b0, acc0);
            acc1 = WMMA_F32(a0, b1, acc1);
            acc2 = WMMA_F32(a0, b2, acc2);
            acc3 = WMMA_F32(a0, b3, acc3);
            a0 = a1;
        }
        for (int v = 0; v < 8; ++v) {
            float* drow = D + (size_t)(mt * 16 + v + 8 * half) * NC + nbase + lr;
            drow[0]  = acc0[v];
            drow[16] = acc1[v];
            drow[32] = acc2[v];
            drow[48] = acc3[v];
        }
    }
}

// --------------------------------------------------- GCN conv: self-loop init
__global__ void k_selfloop(const float* __restrict__ h, const float* __restrict__ dinv,
                           const float* __restrict__ b, float* __restrict__ agg, int F) {
    size_t idx = (size_t)blockIdx.x * blockDim.x + threadIdx.x;
    if (idx >= (size_t)NN * F) return;
    int i = (int)(idx / F), f = (int)(idx % F);
    float dv = dinv[i];
    agg[idx] = h[idx] * dv * dv + b[f];
}

// ------------------------------------------------------ GCN conv: edge scatter
__global__ void k_scatter(const float* __restrict__ h, const int* __restrict__ row,
                          const int* __restrict__ col, const float* __restrict__ ew,
                          const float* __restrict__ dinv, float* __restrict__ agg, int F) {
    int e    = (blockIdx.x * blockDim.x + threadIdx.x) >> 5;
    int lane = threadIdx.x & 31;
    if (e >= EE) return;
    int r = row[e], c = col[e];
    float nrm = dinv[r] * ew[e] * dinv[c];
    const float* hs = h + (size_t)r * F;
    float* ad = agg + (size_t)c * F;
    for (int f = lane; f < F; f += 32) atomicAdd(ad + f, hs[f] * nrm);
}

__global__ void k_relu(float* __restrict__ x, size_t n) {
    size_t i = (size_t)blockIdx.x * blockDim.x + threadIdx.x;
    if (i < n) x[i] = fmaxf(x[i], 0.f);
}

// -------------------------------------------------------------- BN statistics
__global__ void k_col_stats(const float* __restrict__ X, int rows, int cols,
                            float* __restrict__ meanv, float* __restrict__ varv) {
    __shared__ float ss[256], sq[256];
    int c = blockIdx.x, tid = threadIdx.x;
    float s = 0.f, q = 0.f;
    for (int i = tid; i < rows; i += 256) {
        float v = X[(size_t)i * cols + c];
        s += v; q += v * v;
    }
    ss[tid] = s; sq[tid] = q; __syncthreads();
    for (int st = 128; st > 0; st >>= 1) {
        if (tid < st) { ss[tid] += ss[tid + st]; sq[tid] += sq[tid + st]; }
        __syncthreads();
    }
    if (tid == 0) {
        float m = ss[0] / (float)rows;
        meanv[c] = m;
        varv[c]  = sq[0] / (float)rows - m * m;
    }
}

__global__ void k_bn_apply(float* __restrict__ X, const float* __restrict__ stats,
                           const float* __restrict__ g, const float* __restrict__ b) {
    size_t idx = (size_t)blockIdx.x * blockDim.x + threadIdx.x;
    if (idx >= (size_t)NN * D_OUT) return;
    int c = (int)(idx & (D_OUT - 1));
    float v = X[idx];
    X[idx] = fmaxf(g[c] * (v - stats[c]) * rsqrtf(stats[128 + c] + EPSV) + b[c], 0.f);
}

__global__ void k_edge_stats(const float* __restrict__ ed, float* __restrict__ out2) {
    __shared__ float ss[256], sq[256];
    int tid = threadIdx.x;
    float s = 0.f, q = 0.f;
    for (int i = tid; i < NN; i += 256) { float v = ed[i]; s += v; q += v * v; }
    ss[tid] = s; sq[tid] = q; __syncthreads();
    for (int st = 128; st > 0; st >>= 1) {
        if (tid < st) { ss[tid] += ss[tid + st]; sq[tid] += sq[tid + st]; }
        __syncthreads();
    }
    if (tid == 0) {
        float m = ss[0] / (float)NN;
        out2[0] = m;
        out2[1] = sq[0] / (float)NN - m * m;
    }
}

__global__ void k_e_expand(const float* __restrict__ fc0W, const float* __restrict__ fc0b,
                           float* __restrict__ stats) {
    int f = threadIdx.x;  // 128
    float me = stats[768], ve = stats[769];
    stats[256 + f] = fc0W[f] * me + fc0b[f];
    stats[384 + f] = fc0W[f] * fc0W[f] * ve;
}

__global__ void k_hist(const int* __restrict__ degree, int* __restrict__ hist) {
    int i = blockIdx.x * blockDim.x + threadIdx.x;
    if (i < NN) atomicAdd(&hist[degree[i]], 1);
}

__global__ void k_d_stats(const float* __restrict__ emb, const int* __restrict__ hist,
                          float* __restrict__ stats) {
    int f = threadIdx.x;  // 128
    float s = 0.f, q = 0.f;
    for (int d = 0; d < 500; ++d) {
        float cnt = (float)hist[d];
        float v = emb[d * 128 + f];
        s += cnt * v; q += cnt * v * v;
    }
    float m = s / (float)NN;
    stats[512 + f] = m;
    stats[640 + f] = q / (float)NN - m * m;
}

// ------------------------------------------- fc2 GEMM with "virtual concat" A
__device__ __forceinline__ v2f fc2_afrag(int k0, const float* __restrict__ hrow,
                                         float ed, const float* __restrict__ erow,
                                         const float* __restrict__ fc0W,
                                         const float* __restrict__ fc0b,
                                         const float* __restrict__ beg,
                                         const float* __restrict__ beb,
                                         const float* __restrict__ bdg,
                                         const float* __restrict__ bdb,
                                         const float* __restrict__ stats) {
    v2f a;
    if (k0 < 128) {                        // h-branch (materialized, post BN+relu)
        a = *(const v2f*)(hrow + k0);
    } else if (k0 < 256) {                 // e-branch computed inline
        int f = k0 - 128;
        float x0 = ed * fc0W[f]     + fc0b[f];
        float x1 = ed * fc0W[f + 1] + fc0b[f + 1];
        a.x = fmaxf(beg[f]   * (x0 - stats[256+f])   * rsqrtf(stats[384+f]   + EPSV) + beb[f],   0.f);
        a.y = fmaxf(beg[f+1] * (x1 - stats[256+f+1]) * rsqrtf(stats[384+f+1] + EPSV) + beb[f+1], 0.f);
    } else {                               // d-branch computed inline (emb gather)
        int f = k0 - 256;
        float x0 = erow[f], x1 = erow[f + 1];
        a.x = fmaxf(bdg[f]   * (x0 - stats[512+f])   * rsqrtf(stats[640+f]   + EPSV) + bdb[f],   0.f);
        a.y = fmaxf(bdg[f+1] * (x1 - stats[512+f+1]) * rsqrtf(stats[640+f+1] + EPSV) + bdb[f+1], 0.f);
    }
    return a;
}

__global__ void k_gemm_fc2(const float* __restrict__ hbn, const float* __restrict__ edgev,
                           const int* __restrict__ degree, const float* __restrict__ emb,
                           const float* __restrict__ fc0W, const float* __restrict__ fc0b,
                           const float* __restrict__ beg, const float* __restrict__ beb,
                           const float* __restrict__ bdg, const float* __restrict__ bdb,
                           const float* __restrict__ W, const float* __restrict__ bias,
                           const float* __restrict__ stats, float* __restrict__ D) {
    extern __shared__ float smem[];             // [64][388]
    const int nchunk = blockIdx.x & 1;
    const int msuper = blockIdx.x >> 1;
    const int waveid = threadIdx.x >> 5;
    const int lane   = threadIdx.x & 31;
    const int mt     = msuper * 8 + waveid;
    const int nbase  = nchunk * 64;
    const int ldW    = 388;

    for (int r = waveid; r < 64; r += 8) {
        const float4* src = (const float4*)(W + (size_t)(nbase + r) * 384);
        float* dst = smem + r * ldW;
        for (int c4 = lane; c4 < 96; c4 += 32)
            *(float4*)(dst + c4 * 4) = src[c4];
    }
    __syncthreads();

    if (mt < NN / 16) {
        const int half = lane >> 4, lr = lane & 15;
        const int k2 = 2 * half;
        const int m = mt * 16 + lr;
        const float ed = edgev[m];
        const float* erow = emb + (size_t)degree[m] * 128;
        const float* hrow = hbn + (size_t)m * 128;
        const float* s0 = smem + ( 0 + lr) * ldW + k2;
        const float* s1 = smem + (16 + lr) * ldW + k2;
        const float* s2 = smem + (32 + lr) * ldW + k2;
        const float* s3 = smem + (48 + lr) * ldW + k2;

        v2f a = fc2_afrag(k2, hrow, ed, erow, fc0W, fc0b, beg, beb, bdg, bdb, stats);
        v8f acc0 = {}, acc1 = {}, acc2 = {}, acc3 = {};
        for (int k = 0; k < 380; k += 4) {
            v2f an = fc2_afrag(k + 4 + k2, hrow, ed, erow, fc0W, fc0b, beg, beb, bdg, bdb, stats);
            v2f b0 = *(const v2f*)(s0 + k);
            v2f b1 = *(const v2f*)(s1 + k);
            v2f b2 = *(const v2f*)(s2 + k);
            v2f b3 = *(const v2f*)(s3 + k);
            acc0 = WMMA_F32(a, b0, acc0);
            acc1 = WMMA_F32(a, b1, acc1);
            acc2 = WMMA_F32(a, b2, acc2);
            acc3 = WMMA_F32(a, b3, acc3);
            a = an;
        }
        {
            v2f b0 = *(const v2f*)(s0 + 380);
            v2f b1 = *(const v2f*)(s1 + 380);
            v2f b2 = *(const v2f*)(s2 + 380);
            v2f b3 = *(const v2f*)(s3 + 380);
            acc0 = WMMA_F32(a, b0, acc0);
            acc1 = WMMA_F32(a, b1, acc1);
            acc2 = WMMA_F32(a, b2, acc2);
            acc3 = WMMA_F32(a, b3, acc3);
        }
        for (int v = 0; v < 8; ++v) {
            int c0 = nbase + lr;
            float* drow = D + (size_t)(mt * 16 + v + 8 * half) * 128 + c0;
            drow[0]  = acc0[v] + bias[c0];
            drow[16] = acc1[v] + bias[c0 + 16];
            drow[32] = acc2[v] + bias[c0 + 32];
            drow[48] = acc3[v] + bias[c0 + 48];
        }
    }
}

// --------------------------------------------------------------- row L2-norm
__global__ void k_l2norm(const float* __restrict__ src, float* __restrict__ dst) {
    int wave = (blockIdx.x * blockDim.x + threadIdx.x) >> 5;
    int lane = threadIdx.x & 31;
    if (wave >= NN) return;
    float4 v = ((const float4*)(src + (size_t)wave * 128))[lane];
    float ss = v.x * v.x + v.y * v.y + v.z * v.z + v.w * v.w;
    for (int m = 16; m >= 1; m >>= 1) ss += __shfl_xor(ss, m, 32);
    float sc = 1.0f / fmaxf(sqrtf(ss), 1e-12f);
    float4 o = make_float4(v.x * sc, v.y * sc, v.z * sc, v.w * sc);
    ((float4*)(dst + (size_t)wave * 128))[lane] = o;
}

// ----------------------------------------------------- logits + labels output
__global__ void k_logits(const float* __restrict__ embs, const float* __restrict__ queue,
                         const int* __restrict__ idxp, float* __restrict__ out) {
    __shared__ float q[128];
    __shared__ float red[256];
    int i = blockIdx.x, tid = threadIdx.x;
    int qrow = idxp[0] * 256 + i;
    if (tid < 128) q[tid] = embs[(size_t)qrow * 128 + tid];
    __syncthreads();
    float s = 0.f;
    for (int k = 0; k < 128; ++k) s += q[k] * queue[k * 256 + tid];
    const size_t base = (size_t)NN * 128;
    out[base + (size_t)i * 257 + 1 + tid] = s / 0.07f;
    red[tid] = (tid < 128) ? q[tid] * q[tid] : 0.f;
    __syncthreads();
    for (int st = 128; st > 0; st >>= 1) {
        if (tid < st) red[tid] += red[tid + st];
        __syncthreads();
    }
    if (tid == 0) {
        out[base + (size_t)i * 257] = red[0] / 0.07f;      // l_pos
        out[base + (size_t)256 * 257 + i] = 0.0f;          // label (zero bits == int 0)
    }
}

// ------------------------------------------------------------------ launcher
extern "C" void kernel_launch(void* const* d_in, const int* in_sizes, int n_in,
                              void* d_out, int out_size, void* d_ws, size_t ws_size,
                              hipStream_t stream) {
    (void)in_sizes; (void)n_in; (void)out_size; (void)ws_size;
    const float* x      = (const float*)d_in[0];
    const int*   ei     = (const int*)d_in[1];
    const int*   erow   = ei;
    const int*   ecol   = ei + EE;
    const float* ew     = (const float*)d_in[2];
    const float* edgev  = (const float*)d_in[3];
    const int*   degree = (const int*)d_in[4];
    const int*   idxp   = (const int*)d_in[5];
    const float* W1     = (const float*)d_in[7];
    const float* b1     = (const float*)d_in[8];
    const float* W2     = (const float*)d_in[9];
    const float* b2     = (const float*)d_in[10];
    const float* bn_g   = (const float*)d_in[11];
    const float* bn_b   = (const float*)d_in[12];
    const float* be_g   = (const float*)d_in[13];
    const float* be_b   = (const float*)d_in[14];
    const float* bd_g   = (const float*)d_in[15];
    const float* bd_b   = (const float*)d_in[16];
    const float* fc0_W  = (const float*)d_in[17];
    const float* fc0_b  = (const float*)d_in[18];
    const float* emb    = (const float*)d_in[19];
    const float* fc2_W  = (const float*)d_in[20];
    const float* fc2_b  = (const float*)d_in[21];
    const float* queue  = (const float*)d_in[22];
    float* out = (float*)d_out;

    // workspace layout (region A is reused: h1 -> h2 / out_pre)
    float* h1    = (float*)d_ws;                       // N*256
    float* agg1  = h1   + (size_t)NN * D_H;            // N*256
    float* agg2  = agg1 + (size_t)NN * D_H;            // N*128 (becomes hbn in place)
    float* deg   = agg2 + (size_t)NN * D_OUT;          // N
    float* dinv  = deg  + NN;                          // N
    float* stats = dinv + NN;                          // 1024 floats
    int*   hist  = (int*)(stats + 1024);               // 512 ints
    float* h2      = h1;                               // N*128 (h1 dead by GEMM2)
    float* out_pre = h1 + (size_t)NN * D_OUT;          // N*128

    const int T = 256;
    const int Mtiles = NN / 16;                        // 3125 (exact)
    const int msupers = (Mtiles + 7) / 8;              // 391

    // degrees
    k_fill_f32<<<(NN + T - 1) / T, T, 0, stream>>>(deg, 1.0f, NN);
    k_deg_scatter<<<(EE + T - 1) / T, T, 0, stream>>>(ecol, ew, deg);
    k_dinv<<<(NN + T - 1) / T, T, 0, stream>>>(deg, dinv);

    // conv1: h1 = x @ W1^T ; agg1 = selfloop + scatter + b1 ; relu
    k_gemm_wmma<<<msupers * (D_H / 64), T, 64 * (D_IN + 4) * sizeof(float), stream>>>(
        x, W1, h1, Mtiles, D_IN, D_H);
    k_selfloop<<<(int)(((size_t)NN * D_H + T - 1) / T), T, 0, stream>>>(h1, dinv, b1, agg1, D_H);
    k_scatter<<<(int)(((size_t)EE * 32 + T - 1) / T), T, 0, stream>>>(h1, erow, ecol, ew, dinv, agg1, D_H);
    k_relu<<<(int)(((size_t)NN * D_H + T - 1) / T), T, 0, stream>>>(agg1, (size_t)NN * D_H);

    // conv2: h2 = agg1 @ W2^T ; agg2 = selfloop + scatter + b2
    k_gemm_wmma<<<msupers * (D_OUT / 64), T, 64 * (D_H + 4) * sizeof(float), stream>>>(
        agg1, W2, h2, Mtiles, D_H, D_OUT);
    k_selfloop<<<(int)(((size_t)NN * D_OUT + T - 1) / T), T, 0, stream>>>(h2, dinv, b2, agg2, D_OUT);
    k_scatter<<<(int)(((size_t)EE * 32 + T - 1) / T), T, 0, stream>>>(h2, erow, ecol, ew, dinv, agg2, D_OUT);

    // BN(h) + relu (in place)
    k_col_stats<<<128, T, 0, stream>>>(agg2, NN, D_OUT, stats, stats + 128);
    k_bn_apply<<<(int)(((size_t)NN * D_OUT + T - 1) / T), T, 0, stream>>>(agg2, stats, bn_g, bn_b);

    // e-branch / d-branch stats (analytic)
    k_edge_stats<<<1, T, 0, stream>>>(edgev, stats + 768);
    k_e_expand<<<1, 128, 0, stream>>>(fc0_W, fc0_b, stats);
    k_zero_i32<<<1, 512, 0, stream>>>(hist, 512);
    k_hist<<<(NN + T - 1) / T, T, 0, stream>>>(degree, hist);
    k_d_stats<<<1, 128, 0, stream>>>(emb, hist, stats);

    // fc2 with virtual concat -> out_pre
    k_gemm_fc2<<<msupers * 2, T, 64 * 388 * sizeof(float), stream>>>(
        agg2, edgev, degree, emb, fc0_W, fc0_b, be_g, be_b, bd_g, bd_b,
        fc2_W, fc2_b, stats, out_pre);

    // embs_q = l2norm rows -> d_out[0 : N*128]
    k_l2norm<<<(NN * 32 + T - 1) / T, T, 0, stream>>>(out_pre, out);

    // logits + labels -> d_out tail
    k_logits<<<256, T, 0, stream>>>(out, queue, idxp, out);
}